// RESCALBenchmark_88304527606185
// MI455X (gfx1250) — compile-verified
//
#include <hip/hip_runtime.h>

#define NUM_ENTITIES  1000000
#define NUM_RELATIONS 1000
#define RANK          128
#define BATCH         16384
#define LDS_STRIDE    132   // 128 + 4 pad for H/T tiles: conflict-free A-fragment reads

typedef __attribute__((ext_vector_type(2))) float v2f;
typedef __attribute__((ext_vector_type(8))) float v8f;
typedef int v4i __attribute__((vector_size(16)));   // matches async-LDS builtin param type

#define AS3 __attribute__((address_space(3)))

#if defined(__has_builtin)
#if __has_builtin(__builtin_amdgcn_global_load_async_to_lds_b128)
#define HAVE_ASYNC_LDS 1
#endif
#endif

// LDS layout (dynamic): Wl 64KB | Hs 16*132*4 | Ts 16*132*4 | partial 4*16*4
#define WL_F2      (32 * 128 * 2)                 // float2 count: [c][k][kk]
#define HS_OFF     (WL_F2 * 8)                    // bytes
#define TS_OFF     (HS_OFF + 16 * LDS_STRIDE * 4)
#define PART_OFF   (TS_OFF + 16 * LDS_STRIDE * 4)
#define SMEM_BYTES (PART_OFF + 4 * 16 * 4)

// ---------------- grouping pipeline ----------------

__global__ void zero_ws_kernel(int* __restrict__ counts, int* __restrict__ cursors) {
    int t = blockIdx.x * blockDim.x + threadIdx.x;
    if (t < NUM_RELATIONS) { counts[t] = 0; cursors[t] = 0; }
}

__global__ void hist_kernel(const int* __restrict__ rels, int* __restrict__ counts) {
    int b = blockIdx.x * blockDim.x + threadIdx.x;
    if (b < BATCH) atomicAdd(&counts[rels[b]], 1);
}

__global__ __launch_bounds__(1024) void scan_kernel(const int* __restrict__ counts,
                                                    int* __restrict__ offsets) {
    __shared__ int s[1024];
    int t = threadIdx.x;
    s[t] = (t < NUM_RELATIONS) ? counts[t] : 0;
    __syncthreads();
    for (int off = 1; off < 1024; off <<= 1) {
        int v = (t >= off) ? s[t - off] : 0;
        __syncthreads();
        s[t] += v;
        __syncthreads();
    }
    if (t < NUM_RELATIONS) offsets[t + 1] = s[t];
    if (t == 0) offsets[0] = 0;
}

__global__ void scatter_kernel(const int* __restrict__ rels, const int* __restrict__ offsets,
                               int* __restrict__ cursors, int* __restrict__ grouped) {
    int b = blockIdx.x * blockDim.x + threadIdx.x;
    if (b < BATCH) {
        int r = rels[b];
        int pos = offsets[r] + atomicAdd(&cursors[r], 1);
        grouped[pos] = b;
    }
}

// ---------------- main compute: one workgroup (4 waves) per relation ----------------
// W_r staged ONCE into LDS in pair-interleaved layout; per 16-item tile:
// mid(16x128) = H_tile x W_r via v_wmma_f32_16x16x4_f32, epilogue dots with tails.

__global__ __launch_bounds__(128) void rescal_kernel(
    const float* __restrict__ E, const float* __restrict__ W,
    const int* __restrict__ heads, const int* __restrict__ tails,
    const int* __restrict__ offsets, const int* __restrict__ grouped,
    float* __restrict__ out) {

    const int r = blockIdx.x;
    const int start = offsets[r];
    const int count = offsets[r + 1] - start;
    if (count == 0) return;

    extern __shared__ char smem[];
    v2f*   Wl      = (v2f*)smem;                      // [c=32][k=128][kk=2]
    float* Hs      = (float*)(smem + HS_OFF);         // [16][LDS_STRIDE]
    float* Ts      = (float*)(smem + TS_OFF);         // [16][LDS_STRIDE]
    float* partial = (float*)(smem + PART_OFF);       // [4][16]

    const int tid  = threadIdx.x;
    const int lane = tid & 31;
    const int wave = tid >> 5;      // each wave owns k-range [32*wave, 32*wave+32)
    const int lo   = lane & 15;
    const int hi2  = lane >> 4;

    const float* __restrict__ Wr = W + (size_t)r * (RANK * RANK);
    const int col0 = wave * 32 + lo;
    const int col1 = col0 + 16;

    // ---- stage W_r into LDS, pair-interleaved: Wl[c*256 + k*2 + kk] = {W[4c+2kk][k], W[4c+2kk+1][k]}
    #pragma unroll 4
    for (int i = 0; i < 32; ++i) {
        const int e  = tid + 128 * i;     // cell: p = row-pair (d=2p), k0 = col pair base
        const int p  = e >> 6;
        const int k0 = (e & 63) * 2;
        v2f g0 = *(const v2f*)&Wr[(size_t)(2 * p)     * RANK + k0];
        v2f g1 = *(const v2f*)&Wr[(size_t)(2 * p + 1) * RANK + k0];
        const int base = (p >> 1) * 256 + k0 * 2 + (p & 1);
        v2f w0; w0.x = g0.x; w0.y = g1.x;
        v2f w1; w1.x = g0.y; w1.y = g1.y;
        Wl[base]     = w0;
        Wl[base + 2] = w1;
    }

    for (int t0 = 0; t0 < count; t0 += 16) {
        // ---- stage H (heads) and T (tails) tiles: 16 rows x 128 f32 each
        const int m   = tid >> 3;   // 0..15
        const int seg = tid & 7;    // 16 floats per thread
        const int idx = t0 + m;
        if (idx < count) {
            const int b = grouped[start + idx];
            const float* hrow = E + (size_t)heads[b] * RANK + seg * 16;
            const float* trow = E + (size_t)tails[b] * RANK + seg * 16;
#ifdef HAVE_ASYNC_LDS
            #pragma unroll
            for (int j = 0; j < 4; ++j) {
                __builtin_amdgcn_global_load_async_to_lds_b128(
                    (v4i*)(hrow + j * 4),
                    (AS3 v4i*)&Hs[m * LDS_STRIDE + seg * 16 + j * 4], 0, 0);
                __builtin_amdgcn_global_load_async_to_lds_b128(
                    (v4i*)(trow + j * 4),
                    (AS3 v4i*)&Ts[m * LDS_STRIDE + seg * 16 + j * 4], 0, 0);
            }
#else
            #pragma unroll
            for (int j = 0; j < 4; ++j) {
                *(float4*)&Hs[m * LDS_STRIDE + seg * 16 + j * 4] = ((const float4*)hrow)[j];
                *(float4*)&Ts[m * LDS_STRIDE + seg * 16 + j * 4] = ((const float4*)trow)[j];
            }
#endif
        } else {
            const float4 z = make_float4(0.f, 0.f, 0.f, 0.f);
            #pragma unroll
            for (int j = 0; j < 4; ++j) {
                *(float4*)&Hs[m * LDS_STRIDE + seg * 16 + j * 4] = z;
                *(float4*)&Ts[m * LDS_STRIDE + seg * 16 + j * 4] = z;
            }
        }
#ifdef HAVE_ASYNC_LDS
#if __has_builtin(__builtin_amdgcn_s_wait_asynccnt)
        __builtin_amdgcn_s_wait_asynccnt(0);
#else
        asm volatile("s_wait_asynccnt 0x0" ::: "memory");
#endif
#endif
        __syncthreads();

        // ---- GEMM: acc = H_tile x W_r (all operands now in LDS)
        v8f acc0 = {};
        v8f acc1 = {};
        #pragma unroll 4
        for (int c = 0; c < 32; ++c) {
            v2f a  = *(const v2f*)&Hs[lo * LDS_STRIDE + 4 * c + 2 * hi2];
            v2f b0 = Wl[c * 256 + col0 * 2 + hi2];
            v2f b1 = Wl[c * 256 + col1 * 2 + hi2];
            acc0 = __builtin_amdgcn_wmma_f32_16x16x4_f32(
                false, a, false, b0, (short)0, acc0, false, false);
            acc1 = __builtin_amdgcn_wmma_f32_16x16x4_f32(
                false, a, false, b1, (short)0, acc1, false, false);
        }

        // ---- epilogue: score_m += mid[m][k] * T[m][k] over this wave's 32 columns
        #pragma unroll
        for (int rr = 0; rr < 8; ++rr) {
            const int mm = rr + 8 * hi2;     // D layout: vgpr rr -> row rr + 8*(lane/16)
            float v = acc0[rr] * Ts[mm * LDS_STRIDE + col0] +
                      acc1[rr] * Ts[mm * LDS_STRIDE + col1];
            v += __shfl_xor(v, 1, 32);
            v += __shfl_xor(v, 2, 32);
            v += __shfl_xor(v, 4, 32);
            v += __shfl_xor(v, 8, 32);       // stays within each 16-lane half
            if (lo == 0) partial[wave * 16 + mm] = v;
        }
        __syncthreads();

        if (tid < 16) {
            const int idx2 = t0 + tid;
            if (idx2 < count) {
                float s = partial[0 * 16 + tid] + partial[1 * 16 + tid] +
                          partial[2 * 16 + tid] + partial[3 * 16 + tid];
                out[grouped[start + idx2]] = s;
            }
        }
        __syncthreads();
    }
}

// ---------------- launcher ----------------

extern "C" void kernel_launch(void* const* d_in, const int* in_sizes, int n_in,
                              void* d_out, int out_size, void* d_ws, size_t ws_size,
                              hipStream_t stream) {
    const float* E     = (const float*)d_in[0];
    const float* W     = (const float*)d_in[1];
    const int*   heads = (const int*)d_in[2];
    const int*   rels  = (const int*)d_in[3];
    const int*   tails = (const int*)d_in[4];
    float* out = (float*)d_out;

    int* ws      = (int*)d_ws;
    int* counts  = ws;                          // [1000]
    int* offsets = counts + NUM_RELATIONS;      // [1001]
    int* cursors = offsets + NUM_RELATIONS + 1; // [1000]
    int* grouped = cursors + NUM_RELATIONS;     // [16384]

    zero_ws_kernel<<<(NUM_RELATIONS + 255) / 256, 256, 0, stream>>>(counts, cursors);
    hist_kernel<<<(BATCH + 255) / 256, 256, 0, stream>>>(rels, counts);
    scan_kernel<<<1, 1024, 0, stream>>>(counts, offsets);
    scatter_kernel<<<(BATCH + 255) / 256, 256, 0, stream>>>(rels, offsets, cursors, grouped);
    rescal_kernel<<<NUM_RELATIONS, 128, SMEM_BYTES, stream>>>(
        E, W, heads, tails, offsets, grouped, out);
}